// pos_prediction_57337813401807
// MI455X (gfx1250) — compile-verified
//
#include <hip/hip_runtime.h>
#include <math.h>

#define RR 10000

typedef __attribute__((ext_vector_type(2))) float v2f;
typedef __attribute__((ext_vector_type(8))) float v8f;
typedef __attribute__((ext_vector_type(4))) unsigned int u32x4;
typedef __attribute__((ext_vector_type(8))) int i32x8;
typedef __attribute__((ext_vector_type(4))) int i32x4;

__device__ __forceinline__ v8f wmma4(v2f a, v2f b, v8f c) {
  // D(16x16,f32) = A(16x4,f32) * B(4x16,f32) + C
  return __builtin_amdgcn_wmma_f32_16x16x4_f32(false, a, false, b, (short)0, c,
                                               false, false);
}

__device__ __forceinline__ float sigm(float x) { return 1.f / (1.f + expf(-x)); }

__device__ __forceinline__ void sh9c(float x, float y, float z, float* o) {
  const float s3 = 1.7320508075688772f, s15 = 3.872983346207417f,
              s5 = 2.23606797749979f, inv = 0.28209479177387814f; // 1/sqrt(4pi)
  o[0] = inv;
  o[1] = s3 * y * inv;
  o[2] = s3 * z * inv;
  o[3] = s3 * x * inv;
  o[4] = s15 * x * y * inv;
  o[5] = s15 * y * z * inv;
  o[6] = 0.5f * s5 * (2.f * z * z - x * x - y * y) * inv;
  o[7] = s15 * x * z * inv;
  o[8] = 0.5f * s15 * (x * x - y * y) * inv;
}

// ---- pack a [K,N] f32 weight into WMMA-B fragment order [N/16][K/4][32][2] ----
__global__ __launch_bounds__(256) void k_pack(const float* __restrict__ w,
                                              float* __restrict__ out, int K, int N) {
  const int idx = blockIdx.x * 256 + threadIdx.x;
  const int ksteps = K >> 2;
  const int total = (N >> 4) * ksteps * 32;
  if (idx >= total) return;
  const int lane = idx & 31;
  const int ks = (idx >> 5) % ksteps;
  const int tile = (idx >> 5) / ksteps;
  const int n = tile * 16 + (lane & 15);
  const int k = ks * 4 + ((lane >> 4) << 1);
  out[idx * 2 + 0] = w[k * N + n];
  out[idx * 2 + 1] = w[(k + 1) * N + n];
}

// ---- shared equivariant Linear(480->576) via f32 WMMA with packed B frags ----
__device__ void lin480_wmma(const float xs[][484], float mid[][580],
                            const float* __restrict__ pw0, const float* __restrict__ b0,
                            const float* __restrict__ pw1, const float* __restrict__ pw2,
                            int wave, int lane) {
  const int m_l = lane & 15;
  const int half = lane >> 4;
  const int koff = half * 2;
  // scalars: [16,128]@[128,224] -> mid[:,0:224]
  for (int tile = wave; tile < 14; tile += 4) {
    v8f acc = {0.f, 0.f, 0.f, 0.f, 0.f, 0.f, 0.f, 0.f};
    const int n = tile * 16 + m_l;
    const float* bp = pw0 + (size_t)tile * 32 * 64 + lane * 2;
    for (int ks = 0; ks < 32; ++ks) {
      const int k0 = ks * 4;
      v2f a;
      a.x = xs[m_l][k0 + koff];
      a.y = xs[m_l][k0 + koff + 1];
      const v2f b = *(const v2f*)(bp + ks * 64);
      acc = wmma4(a, b, acc);
    }
    const float bias = b0[n];
#pragma unroll
    for (int v = 0; v < 8; ++v)
      mid[v + half * 8][n] = acc[v] * 0.08838834764831845f + bias; // 1/sqrt(128)
  }
  // vectors: per d, [16,64]@[64,64] -> mid[:,224+k*3+d]
  for (int job = wave; job < 12; job += 4) {
    const int d = job >> 2, tile = job & 3;
    const int n = tile * 16 + m_l;
    v8f acc = {0.f, 0.f, 0.f, 0.f, 0.f, 0.f, 0.f, 0.f};
    const float* bp = pw1 + (size_t)tile * 16 * 64 + lane * 2;
    for (int ks = 0; ks < 16; ++ks) {
      const int k0 = ks * 4;
      v2f a;
      a.x = xs[m_l][128 + (k0 + koff) * 3 + d];
      a.y = xs[m_l][128 + (k0 + koff + 1) * 3 + d];
      const v2f b = *(const v2f*)(bp + ks * 64);
      acc = wmma4(a, b, acc);
    }
#pragma unroll
    for (int v = 0; v < 8; ++v)
      mid[v + half * 8][224 + n * 3 + d] = acc[v] * 0.125f; // 1/sqrt(64)
  }
  // tensors: per d, [16,32]@[32,32] -> mid[:,416+k*5+d]
  for (int job = wave; job < 10; job += 4) {
    const int d = job >> 1, tile = job & 1;
    const int n = tile * 16 + m_l;
    v8f acc = {0.f, 0.f, 0.f, 0.f, 0.f, 0.f, 0.f, 0.f};
    const float* bp = pw2 + (size_t)tile * 8 * 64 + lane * 2;
    for (int ks = 0; ks < 8; ++ks) {
      const int k0 = ks * 4;
      v2f a;
      a.x = xs[m_l][320 + (k0 + koff) * 5 + d];
      a.y = xs[m_l][320 + (k0 + koff + 1) * 5 + d];
      const v2f b = *(const v2f*)(bp + ks * 64);
      acc = wmma4(a, b, acc);
    }
#pragma unroll
    for (int v = 0; v < 8; ++v)
      mid[v + half * 8][416 + n * 5 + d] = acc[v] * 0.17677669529663687f; // 1/sqrt(32)
  }
}

__device__ void gate480(const float mid[][580], float xs[][484], int t) {
  for (int idx = t; idx < 16 * 480; idx += 128) {
    const int nd = idx / 480, c = idx - nd * 480;
    float v;
    if (c < 128) {
      const float u = mid[nd][c];
      v = u * sigm(u);
    } else if (c < 320) {
      const int k = (c - 128) / 3;
      v = mid[nd][224 + (c - 128)] * sigm(mid[nd][128 + k]);
    } else {
      const int k = (c - 320) / 5;
      v = mid[nd][416 + (c - 320)] * sigm(mid[nd][192 + k]);
    }
    xs[nd][c] = v;
  }
}

// ---- K1: gaussian smearing + t_fc + silu ----
__global__ __launch_bounds__(128) void k_graph_prep(const float* __restrict__ mnf,
                                                    const float* __restrict__ target,
                                                    const float* __restrict__ tw,
                                                    const float* __restrict__ tb,
                                                    float* __restrict__ gadd) {
  __shared__ float gs[64];
  const int g = blockIdx.x, t = threadIdx.x;
  if (t < 64) {
    const float step = 10.f / 63.f;
    const float off = -5.f + step * (float)t;
    const float d = target[g] - off;
    gs[t] = expf((-0.5f / (step * step)) * d * d);
  }
  __syncthreads();
  float acc = 0.f;
  for (int k = 0; k < 64; ++k) acc += gs[k] * tw[k * 128 + t];
  acc = acc * 0.125f + tb[t]; // 1/sqrt(64)
  const float se = acc * sigm(acc);
  gadd[g * 480 + t] = mnf[g * 480 + t] + se;
  for (int c = 128 + t; c < 480; c += 128) gadd[g * 480 + c] = mnf[g * 480 + c];
}

// ---- K1b: SH grid [10000][12] (zero K-pad) ----
__global__ void k_grid(float* __restrict__ gridp) {
  const int r = blockIdx.x * blockDim.x + threadIdx.x;
  if (r >= RR) return;
  const int i = r / 100, j = r - i * 100;
  const float b = ((float)i + 0.5f) * (3.14159265358979323846f / 100.f);
  const float a = (float)j * (6.283185307179586f / 100.f);
  const float sb = sinf(b), cb = cosf(b);
  const float x = sb * cosf(a), y = sb * sinf(a), z = cb;
  float o[9];
  sh9c(x, y, z, o);
  float* d = gridp + (size_t)r * 12;
#pragma unroll
  for (int q = 0; q < 9; ++q) d[q] = o[q];
  d[9] = d[10] = d[11] = 0.f;
}

// ---- K1c: grid into WMMA-B fragment order [625 r-tiles][3 ksteps][32][2] ----
__global__ __launch_bounds__(256) void k_gridpack(const float* __restrict__ gridp,
                                                  float* __restrict__ gf) {
  const int idx = blockIdx.x * 256 + threadIdx.x;
  if (idx >= 625 * 3 * 32) return;
  const int lane = idx & 31;
  const int ks = (idx >> 5) % 3;
  const int rt = (idx >> 5) / 3;
  const int r = rt * 16 + (lane & 15);
  const int k = ks * 4 + ((lane >> 4) << 1);
  gf[idx * 2 + 0] = gridp[(size_t)r * 12 + k];
  gf[idx * 2 + 1] = gridp[(size_t)r * 12 + k + 1];
}

// ---- K2: per-node eq_norm -> WMMA lin480 -> gate -> logit, 16 nodes/block ----
__global__ __launch_bounds__(128) void k_node(
    const float* __restrict__ nf, const int* __restrict__ batch,
    const float* __restrict__ gadd, const float* __restrict__ norm_w,
    const float* __restrict__ norm_b, const float* __restrict__ pw0,
    const float* __restrict__ b0, const float* __restrict__ pw1,
    const float* __restrict__ pw2, const float* __restrict__ f2w,
    const float* __restrict__ f2b, float* __restrict__ ff,
    float* __restrict__ logit) {
  __shared__ float xs[16][484];
  __shared__ float mid[16][580]; // also TDM landing buffer for the nf tile
  __shared__ float red0[16][8], red1[16][8], red2[16][8];
  __shared__ float smean[16], s0[16], s1[16], s2[16];
  __shared__ int bg[16];
  const int t = threadIdx.x;
  const int base = blockIdx.x * 16;
  if (t < 16) bg[t] = batch[base + t];

  // CDNA5 Tensor Data Mover: async 2-D tile copy nf[base:base+16][0:480] -> LDS
  if (t < 32) { // one wave builds the D# and issues the DMA (EXEC ignored)
    const unsigned long long ga =
        (unsigned long long)(uintptr_t)(nf + (size_t)base * 480);
    const unsigned int lds_addr = (unsigned int)(uintptr_t)(&mid[0][0]);
    u32x4 g0;
    i32x8 g1;
    i32x4 g2 = {0, 0, 0, 0}, g3 = {0, 0, 0, 0};
    i32x8 g4 = {0, 0, 0, 0, 0, 0, 0, 0};
    g0[0] = 1u;                                     // count=1 (valid user D#)
    g0[1] = lds_addr;                               // LDS byte address
    g0[2] = (unsigned int)(ga & 0xffffffffu);       // global_addr[31:0]
    g0[3] = (unsigned int)((ga >> 32) & 0x01ffffffu) | 0x80000000u; // addr[56:32]|type=2
    g1[0] = 0x20000;           // workgroup_mask=0, data_size=2 (4 bytes)
    g1[1] = (int)(480u << 16); // tensor_dim0 = 480 (low 16 in [31:16])
    g1[2] = (int)(16u << 16);  // tensor_dim0 hi=0 | tensor_dim1 = 16
    g1[3] = (int)(480u << 16); // tensor_dim1 hi=0 | tile_dim0 = 480
    g1[4] = 16;                // tile_dim1 = 16, tile_dim2 = 0
    g1[5] = 480;               // tensor_dim0_stride (low 32)
    g1[6] = 0;
    g1[7] = 0;
    __builtin_amdgcn_tensor_load_to_lds(g0, g1, g2, g3, g4, 0);
    __builtin_amdgcn_s_wait_tensorcnt(0);
  }
  __syncthreads();

  const float* nfl = (const float*)(&mid[0][0]); // linear [16*480] tile in LDS
  for (int idx = t; idx < 16 * 480; idx += 128) {
    const int nd = idx / 480, c = idx - nd * 480;
    xs[nd][c] = nfl[idx] + gadd[(size_t)bg[nd] * 480 + c];
  }
  __syncthreads();
  // EquivariantLayerNorm
  const int node = t >> 3, j = t & 7;
  {
    float s = 0.f;
    for (int c = j; c < 128; c += 8) s += xs[node][c];
    red0[node][j] = s;
  }
  __syncthreads();
  if (j == 0) {
    float m = 0.f;
    for (int q = 0; q < 8; ++q) m += red0[node][q];
    smean[node] = m * (1.f / 128.f);
  }
  __syncthreads();
  {
    const float mn = smean[node];
    float a = 0.f, b = 0.f, c2 = 0.f;
    for (int c = j; c < 128; c += 8) {
      const float v = xs[node][c] - mn;
      xs[node][c] = v;
      a += v * v;
    }
    for (int c = 128 + j; c < 320; c += 8) {
      const float v = xs[node][c];
      b += v * v;
    }
    for (int c = 320 + j; c < 480; c += 8) {
      const float v = xs[node][c];
      c2 += v * v;
    }
    red0[node][j] = a;
    red1[node][j] = b;
    red2[node][j] = c2;
  }
  __syncthreads();
  if (j == 0) {
    float a = 0.f, b = 0.f, c2 = 0.f;
    for (int q = 0; q < 8; ++q) {
      a += red0[node][q];
      b += red1[node][q];
      c2 += red2[node][q];
    }
    s0[node] = rsqrtf(a * (1.f / 128.f) + 1e-5f);
    s1[node] = rsqrtf(b * (1.f / 192.f) + 1e-5f);
    s2[node] = rsqrtf(c2 * (1.f / 160.f) + 1e-5f);
  }
  __syncthreads();
  for (int idx = t; idx < 16 * 480; idx += 128) {
    const int nd = idx / 480, c = idx - nd * 480;
    float v = xs[nd][c];
    if (c < 128)
      v = v * s0[nd] * norm_w[c] + norm_b[c];
    else if (c < 320) {
      const int m = (c - 128) / 3;
      v = v * s1[nd] * norm_w[128 + m];
    } else {
      const int m = (c - 320) / 5;
      v = v * s2[nd] * norm_w[192 + m];
    }
    xs[nd][c] = v;
  }
  __syncthreads();
  lin480_wmma(xs, mid, pw0, b0, pw1, pw2, t >> 5, t & 31);
  __syncthreads();
  gate480(mid, xs, t);
  __syncthreads();
  // focus logit: ff[:, :128] @ f2_w0 / sqrt(128) + b
  {
    float p = 0.f;
    for (int c = j; c < 128; c += 8) p += xs[node][c] * f2w[c];
    red0[node][j] = p;
  }
  __syncthreads();
  if (j == 0) {
    float s = 0.f;
    for (int q = 0; q < 8; ++q) s += red0[node][q];
    logit[base + node] = s * 0.08838834764831845f + f2b[0];
  }
  for (int idx = t; idx < 16 * 480; idx += 128) {
    const int nd = idx / 480, c = idx - nd * 480;
    ff[(size_t)(base + nd) * 480 + c] = xs[nd][c];
  }
}

// ---- K3: segment softmax over 32 nodes/graph, agg + focus-weighted center ----
__global__ __launch_bounds__(128) void k_agg(const float* __restrict__ pos,
                                             const float* __restrict__ ff,
                                             const float* __restrict__ logit,
                                             float* __restrict__ agg,
                                             float* __restrict__ center) {
  __shared__ float fo[32];
  __shared__ float stat;
  const int g = blockIdx.x, t = threadIdx.x, n0 = g * 32;
  if (t < 32) fo[t] = logit[n0 + t];
  __syncthreads();
  if (t == 0) {
    float m = fo[0];
    for (int q = 1; q < 32; ++q) m = fmaxf(m, fo[q]);
    float s = 0.f;
    for (int q = 0; q < 32; ++q) {
      fo[q] = expf(fo[q] - m);
      s += fo[q];
    }
    stat = 1.f / s;
  }
  __syncthreads();
  const float inv = stat;
  for (int c = t; c < 480; c += 128) {
    float a = 0.f;
    for (int q = 0; q < 32; ++q) a += fo[q] * ff[(size_t)(n0 + q) * 480 + c];
    agg[(size_t)g * 480 + c] = a * inv;
  }
  if (t < 3) {
    float a = 0.f;
    for (int q = 0; q < 32; ++q) a += fo[q] * pos[(n0 + q) * 3 + t];
    center[g * 4 + t] = a * inv;
  }
}

// ---- K4: position head -> pos_out[G][16][9] ----
__global__ __launch_bounds__(128) void k_poshead(
    const float* __restrict__ agg, const float* __restrict__ pw0,
    const float* __restrict__ b0, const float* __restrict__ pw1,
    const float* __restrict__ pw2, const float* __restrict__ pq0,
    const float* __restrict__ qb, const float* __restrict__ pq1,
    const float* __restrict__ pq2, float* __restrict__ posout) {
  __shared__ float xs[16][484];
  __shared__ float mid[16][580];
  const int t = threadIdx.x;
  const int g0 = blockIdx.x * 16;
  for (int idx = t; idx < 16 * 480; idx += 128) {
    const int nd = idx / 480, c = idx - nd * 480;
    xs[nd][c] = agg[(size_t)(g0 + nd) * 480 + c];
  }
  __syncthreads();
  lin480_wmma(xs, mid, pw0, b0, pw1, pw2, t >> 5, t & 31);
  __syncthreads();
  gate480(mid, xs, t);
  __syncthreads();
  const int lane = t & 31, wave = t >> 5, m_l = lane & 15, half = lane >> 4,
            koff = half * 2;
  for (int job = wave; job < 9; job += 4) {
    v8f acc = {0.f, 0.f, 0.f, 0.f, 0.f, 0.f, 0.f, 0.f};
    const int n = m_l;
    if (job == 0) { // o0: [16,128]@[128,16]
      const float* bp = pq0 + lane * 2;
      for (int ks = 0; ks < 32; ++ks) {
        const int k0 = ks * 4;
        v2f a;
        a.x = xs[m_l][k0 + koff];
        a.y = xs[m_l][k0 + koff + 1];
        const v2f b = *(const v2f*)(bp + ks * 64);
        acc = wmma4(a, b, acc);
      }
      const float bias = qb[n];
#pragma unroll
      for (int v = 0; v < 8; ++v)
        posout[(size_t)(g0 + v + half * 8) * 144 + n * 9 + 0] =
            acc[v] * 0.08838834764831845f + bias;
    } else if (job < 4) { // o1 per d: [16,64]@[64,16]
      const int d = job - 1;
      const float* bp = pq1 + lane * 2;
      for (int ks = 0; ks < 16; ++ks) {
        const int k0 = ks * 4;
        v2f a;
        a.x = xs[m_l][128 + (k0 + koff) * 3 + d];
        a.y = xs[m_l][128 + (k0 + koff + 1) * 3 + d];
        const v2f b = *(const v2f*)(bp + ks * 64);
        acc = wmma4(a, b, acc);
      }
#pragma unroll
      for (int v = 0; v < 8; ++v)
        posout[(size_t)(g0 + v + half * 8) * 144 + n * 9 + 1 + d] = acc[v] * 0.125f;
    } else { // o2 per d: [16,32]@[32,16]
      const int d = job - 4;
      const float* bp = pq2 + lane * 2;
      for (int ks = 0; ks < 8; ++ks) {
        const int k0 = ks * 4;
        v2f a;
        a.x = xs[m_l][320 + (k0 + koff) * 5 + d];
        a.y = xs[m_l][320 + (k0 + koff + 1) * 5 + d];
        const v2f b = *(const v2f*)(bp + ks * 64);
        acc = wmma4(a, b, acc);
      }
#pragma unroll
      for (int v = 0; v < 8; ++v)
        posout[(size_t)(g0 + v + half * 8) * 144 + n * 9 + 4 + d] =
            acc[v] * 0.17677669529663687f;
    }
  }
}

// ---- K5: fused grid einsum (WMMA, packed B) + channel-LSE + r-softmax ----
__global__ __launch_bounds__(128) void k_resout(const float* __restrict__ posout,
                                                const float* __restrict__ gf,
                                                float* __restrict__ res) {
  __shared__ float A[16][12];
  __shared__ float pl[RR];
  __shared__ float wred[128];
  __shared__ float gstat[2];
  const int g = blockIdx.x, t = threadIdx.x;
  for (int idx = t; idx < 192; idx += 128) {
    const int c = idx / 12, i = idx - c * 12;
    A[c][i] = (i < 9) ? posout[(size_t)g * 144 + c * 9 + i] : 0.f;
  }
  __syncthreads();
  const int lane = t & 31, wave = t >> 5, m_l = lane & 15, half = lane >> 4,
            koff = half * 2;
  float lmax = -3.4e38f;
  for (int rt = wave; rt < RR / 16; rt += 4) {
    const int r0 = rt * 16;
    const float* bp = gf + (size_t)rt * 3 * 64 + lane * 2;
    v8f acc = {0.f, 0.f, 0.f, 0.f, 0.f, 0.f, 0.f, 0.f};
#pragma unroll
    for (int ks = 0; ks < 3; ++ks) {
      const int k0 = ks * 4;
      v2f a;
      a.x = A[m_l][k0 + koff];
      a.y = A[m_l][k0 + koff + 1];
      const v2f b = *(const v2f*)(bp + ks * 64);
      acc = wmma4(a, b, acc);
    }
    // logsumexp over the 16 channels (M dim): 8 rows here + 8 in partner lane
    float cm = acc[0];
#pragma unroll
    for (int v = 1; v < 8; ++v) cm = fmaxf(cm, acc[v]);
    cm = fmaxf(cm, __shfl_xor(cm, 16, 32));
    float s = 0.f;
#pragma unroll
    for (int v = 0; v < 8; ++v) s += expf(acc[v] - cm);
    const float tot = s + __shfl_xor(s, 16, 32);
    const float lse = logf(tot); // reference: max subtracted, not added back
    if (half == 0) pl[r0 + m_l] = lse;
    lmax = fmaxf(lmax, lse);
  }
  wred[t] = lmax;
  __syncthreads();
  if (t == 0) {
    float m = wred[0];
    for (int q = 1; q < 128; ++q) m = fmaxf(m, wred[q]);
    gstat[0] = m;
  }
  __syncthreads();
  const float gmax = gstat[0];
  float p = 0.f;
  for (int r = t; r < RR; r += 128) p += expf((pl[r] - gmax) * 100.f);
  wred[t] = p;
  __syncthreads();
  if (t == 0) {
    float s = 0.f;
    for (int q = 0; q < 128; ++q) s += wred[q];
    gstat[1] = 1.f / s;
  }
  __syncthreads();
  const float inv = gstat[1];
  for (int r = t; r < RR; r += 128)
    res[(size_t)g * RR + r] = expf((pl[r] - gmax) * 100.f) * inv;
}

// ---- K6: label path ----
__global__ __launch_bounds__(128) void k_label(const float* __restrict__ mp,
                                               const float* __restrict__ center,
                                               const float* __restrict__ gridp,
                                               float* __restrict__ outl,
                                               float* __restrict__ outlen) {
  __shared__ float sh[12];
  __shared__ float pl[RR];
  __shared__ float wred[128];
  __shared__ float gstat[2];
  const int g = blockIdx.x, t = threadIdx.x;
  if (t == 0) {
    const float x = mp[g * 3 + 0] - center[g * 4 + 0];
    const float y = mp[g * 3 + 1] - center[g * 4 + 1];
    const float z = mp[g * 3 + 2] - center[g * 4 + 2];
    outlen[g] = sqrtf(x * x + y * y + z * z);
    float o[9];
    sh9c(x, y, z, o);
    for (int q = 0; q < 9; ++q) sh[q] = o[q];
    sh[9] = sh[10] = sh[11] = 0.f;
  }
  __syncthreads();
  float lmax = -3.4e38f;
  for (int r = t; r < RR; r += 128) {
    const float* gr = gridp + (size_t)r * 12;
    float v = sh[0] * gr[0] + sh[1] * gr[1] + sh[2] * gr[2] + sh[3] * gr[3] +
              sh[4] * gr[4] + sh[5] * gr[5] + sh[6] * gr[6] + sh[7] * gr[7] +
              sh[8] * gr[8];
    v *= 100.f; // /TEMP
    pl[r] = v;
    lmax = fmaxf(lmax, v);
  }
  wred[t] = lmax;
  __syncthreads();
  if (t == 0) {
    float m = wred[0];
    for (int q = 1; q < 128; ++q) m = fmaxf(m, wred[q]);
    gstat[0] = m;
  }
  __syncthreads();
  const float gmax = gstat[0];
  float p = 0.f;
  for (int r = t; r < RR; r += 128) p += expf(pl[r] - gmax);
  wred[t] = p;
  __syncthreads();
  if (t == 0) {
    float s = 0.f;
    for (int q = 0; q < 128; ++q) s += wred[q];
    gstat[1] = 1.f / s;
  }
  __syncthreads();
  const float inv = gstat[1];
  for (int r = t; r < RR; r += 128)
    outl[(size_t)g * RR + r] = expf(pl[r] - gmax) * inv;
}

extern "C" void kernel_launch(void* const* d_in, const int* in_sizes, int n_in,
                              void* d_out, int out_size, void* d_ws, size_t ws_size,
                              hipStream_t stream) {
  (void)in_sizes; (void)n_in; (void)out_size; (void)ws_size;
  const float* pos           = (const float*)d_in[0];
  const float* node_features = (const float*)d_in[1];
  const float* mnf           = (const float*)d_in[2];
  const float* mask_position = (const float*)d_in[3];
  const float* target        = (const float*)d_in[4];
  const int*   batch         = (const int*)d_in[6]; // d_in[5]=mask_atom unused
  const float* t_fc_w = (const float*)d_in[7];
  const float* t_fc_b = (const float*)d_in[8];
  const float* norm_w = (const float*)d_in[9];
  const float* norm_b = (const float*)d_in[10];
  const float* f1_w0 = (const float*)d_in[11];
  const float* f1_b0 = (const float*)d_in[12];
  const float* f1_w1 = (const float*)d_in[13];
  const float* f1_w2 = (const float*)d_in[14];
  const float* f2_w0 = (const float*)d_in[15];
  const float* f2_b0 = (const float*)d_in[16];
  const float* p1_w0 = (const float*)d_in[17];
  const float* p1_b0 = (const float*)d_in[18];
  const float* p1_w1 = (const float*)d_in[19];
  const float* p1_w2 = (const float*)d_in[20];
  const float* p2_w0 = (const float*)d_in[21];
  const float* p2_b0 = (const float*)d_in[22];
  const float* p2_w1 = (const float*)d_in[23];
  const float* p2_w2 = (const float*)d_in[24];
  float* out = (float*)d_out;
  float* ws = (float*)d_ws;
  // workspace layout (floats)
  float* gadd   = ws + 0;         // 1024*480
  float* ff     = ws + 491520;    // 32768*480
  float* logit  = ws + 16220160;  // 32768
  float* agg    = ws + 16252928;  // 1024*480
  float* center = ws + 16744448;  // 1024*4
  float* posout = ws + 16748544;  // 1024*144
  float* gridp  = ws + 16896000;  // 10000*12
  float* pf1w0  = ws + 17016000;  // 14*32*64
  float* pf1w1  = ws + 17044672;  // 4*16*64
  float* pf1w2  = ws + 17048768;  // 2*8*64
  float* pp1w0  = ws + 17049792;  // 14*32*64
  float* pp1w1  = ws + 17078464;  // 4*16*64
  float* pp1w2  = ws + 17082560;  // 2*8*64
  float* pq0    = ws + 17083584;  // 1*32*64
  float* pq1    = ws + 17085632;  // 1*16*64
  float* pq2    = ws + 17086656;  // 1*8*64
  float* gfrag  = ws + 17087168;  // 625*3*64

  // prep: smearing, SH grid, fragment packing
  k_graph_prep<<<1024, 128, 0, stream>>>(mnf, target, t_fc_w, t_fc_b, gadd);
  k_grid<<<(RR + 127) / 128, 128, 0, stream>>>(gridp);
  k_gridpack<<<(625 * 3 * 32 + 255) / 256, 256, 0, stream>>>(gridp, gfrag);
  k_pack<<<(14 * 32 * 32 + 255) / 256, 256, 0, stream>>>(f1_w0, pf1w0, 128, 224);
  k_pack<<<(4 * 16 * 32 + 255) / 256, 256, 0, stream>>>(f1_w1, pf1w1, 64, 64);
  k_pack<<<(2 * 8 * 32 + 255) / 256, 256, 0, stream>>>(f1_w2, pf1w2, 32, 32);
  k_pack<<<(14 * 32 * 32 + 255) / 256, 256, 0, stream>>>(p1_w0, pp1w0, 128, 224);
  k_pack<<<(4 * 16 * 32 + 255) / 256, 256, 0, stream>>>(p1_w1, pp1w1, 64, 64);
  k_pack<<<(2 * 8 * 32 + 255) / 256, 256, 0, stream>>>(p1_w2, pp1w2, 32, 32);
  k_pack<<<(1 * 32 * 32 + 255) / 256, 256, 0, stream>>>(p2_w0, pq0, 128, 16);
  k_pack<<<(1 * 16 * 32 + 255) / 256, 256, 0, stream>>>(p2_w1, pq1, 64, 16);
  k_pack<<<(1 * 8 * 32 + 255) / 256, 256, 0, stream>>>(p2_w2, pq2, 32, 16);
  // main pipeline
  k_node<<<32768 / 16, 128, 0, stream>>>(node_features, batch, gadd, norm_w,
                                         norm_b, pf1w0, f1_b0, pf1w1, pf1w2,
                                         f2_w0, f2_b0, ff, logit);
  k_agg<<<1024, 128, 0, stream>>>(pos, ff, logit, agg, center);
  k_poshead<<<1024 / 16, 128, 0, stream>>>(agg, pp1w0, p1_b0, pp1w1, pp1w2,
                                           pq0, p2_b0, pq1, pq2, posout);
  k_resout<<<1024, 128, 0, stream>>>(posout, gfrag, out);
  k_label<<<1024, 128, 0, stream>>>(mask_position, center, gridp,
                                    out + (size_t)1024 * RR,
                                    out + (size_t)2 * 1024 * RR);
}